// TTNLayer_1864015807129
// MI455X (gfx1250) — compile-verified
//
#include <hip/hip_runtime.h>
#include <hip/hip_bf16.h>

// Closed-form TTN circuit evaluation (Heisenberg-picture collapse):
//   out[b][g] = prod_{q=0..7} cos( x[b][8g+q] + phi[g][q] )
// phi[g][q] = sum of selected weight entries (fixed per group).
//
// Memory-bound streaming kernel: 16 MB in / 2 MB out -> ~0.81 us at
// 23.3 TB/s HBM. Data path uses the gfx1250 async DMA engine:
//   global_load_async_to_lds_b128  (ASYNCcnt)  ->  s_wait_asynccnt  ->
//   barrier  ->  ds_load_b128 compute reads.

#define TTN_BLOCK 256        // 8 wave32 waves
#define ROWS_PER_BLOCK 32    // 32 rows * 256 B = 8 KB LDS tile

__global__ __launch_bounds__(TTN_BLOCK) void ttn_closed_form_kernel(
    const float* __restrict__ x,     // (B, 64)
    const float* __restrict__ w,     // (8, 7, 2) flat = 112 floats
    float* __restrict__ out,         // (B, 8)
    int B) {
  __shared__ float tile[ROWS_PER_BLOCK * 64];  // 8 KB x-tile
  __shared__ float phi[64];                    // phi[g*8 + q]

  const int tid = threadIdx.x;
  const long long row0 = (long long)blockIdx.x * ROWS_PER_BLOCK;
  const int rows = (int)min((long long)ROWS_PER_BLOCK, (long long)B - row0);

  // ---- Stage the x tile into LDS via the gfx1250 async DMA path ----
  // Each lane issues 16 B global->LDS copies; 256 lanes x 2 issues = 8 KB.
  {
    const int nchunks = rows * 16;  // one chunk = 16 B (4 floats)
    const char* gbase = (const char*)(x + row0 * 64);
#pragma unroll
    for (int k = 0; k < (ROWS_PER_BLOCK * 16) / TTN_BLOCK; ++k) {
      const int c = tid + k * TTN_BLOCK;
      if (c < nchunks) {
        // Low 32 bits of the flat shared-memory address are the wave-relative
        // LDS byte offset (LDS aperture mapping) -> exactly what VDST wants.
        unsigned lds_off = (unsigned)(uintptr_t)(&tile[c * 4]);
        const char* gp = gbase + (long long)c * 16;
        asm volatile("global_load_async_to_lds_b128 %0, %1, off"
                     :
                     : "v"(lds_off), "v"(gp)
                     : "memory");
      }
    }
  }

  // Speculative L2 prefetch of the next block's tile (global_prefetch_b8);
  // overlaps with this block's DMA.
  if (tid < 2) {
    __builtin_prefetch((const char*)(x + row0 * 64) + 8192 + tid * 4096, 0, 0);
  }

  // ---- Compute per-qubit phases while the DMA is in flight ----
  // phi[q] = wg[q]  (+ wg[8 + (q>>1)] if q odd)  (+ wg[12 + (q>>2)] if q%4==3)
  if (tid < 64) {
    const int g = tid >> 3;
    const int q = tid & 7;
    const float* wg = w + g * 14;
    float p = wg[q];
    if (q & 1) p += wg[8 + (q >> 1)];
    if ((q & 3) == 3) p += wg[12 + (q >> 2)];
    phi[tid] = p;
  }

  // DMA completion (ASYNCcnt -> 0), then make the tile visible block-wide.
  asm volatile("s_wait_asynccnt 0x0" ::: "memory");
  __syncthreads();

  // ---- Evaluate: one thread per (row, group) ----
  const int g = tid & 7;
  const int r = tid >> 3;
  if (row0 + r >= B) return;

  const float* tr = &tile[r * 64 + g * 8];
  const float4 a = *(const float4*)(tr);      // ds_load_b128
  const float4 b = *(const float4*)(tr + 4);  // ds_load_b128
  const float* ph = &phi[g * 8];

  const float f0 = cosf(a.x + ph[0]);
  const float f1 = cosf(a.y + ph[1]);
  const float f2 = cosf(a.z + ph[2]);
  const float f3 = cosf(a.w + ph[3]);
  const float f4 = cosf(b.x + ph[4]);
  const float f5 = cosf(b.y + ph[5]);
  const float f6 = cosf(b.z + ph[6]);
  const float f7 = cosf(b.w + ph[7]);

  out[(row0 + r) * 8 + g] = ((f0 * f1) * (f2 * f3)) * ((f4 * f5) * (f6 * f7));
}

extern "C" void kernel_launch(void* const* d_in, const int* in_sizes, int n_in,
                              void* d_out, int out_size, void* d_ws,
                              size_t ws_size, hipStream_t stream) {
  const float* x = (const float*)d_in[0];  // (B, 64) fp32
  const float* w = (const float*)d_in[1];  // (8, 7, 2) fp32
  // d_in[2] (bias) and d_in[3] (class_bias) are unused by the reference.
  float* out = (float*)d_out;              // (B, 8) fp32

  const int B = in_sizes[0] / 64;
  const int grid = (B + ROWS_PER_BLOCK - 1) / ROWS_PER_BLOCK;

  ttn_closed_form_kernel<<<grid, TTN_BLOCK, 0, stream>>>(x, w, out, B);
}